// SpatialMixBlock_47940424958503
// MI455X (gfx1250) — compile-verified
//
#include <hip/hip_runtime.h>
#include <math.h>

#define NN 50000
#define HH 256
#define EE 800000
#define MT 16            // row tile
#define LDP 260          // padded LDS row stride (floats): 260 % 64 = 4 -> conflict-free frag loads
#define WAVES 16         // 16 waves * 16 cols = 256 output columns

typedef __attribute__((ext_vector_type(2))) float v2f;
typedef __attribute__((ext_vector_type(8))) float v8f;

// ---------------- Kernel 1: per-edge distance weights ----------------
__global__ void edge_weight_kernel(const float* __restrict__ attr,
                                   float* __restrict__ w) {
    int e = blockIdx.x * blockDim.x + threadIdx.x;
    if (e >= EE) return;
    float a0 = attr[3 * e + 0];
    float a1 = attr[3 * e + 1];
    float a2 = attr[3 * e + 2];
    float nrm = sqrtf(a0 * a0 + a1 * a1 + a2 * a2 + 1e-12f);
    float we = expf(-4.0f * nrm);
    w[e] = fmaxf(we, 0.0f);
}

// ---------------- Kernel 2: gather + weighted scatter-add ----------------
// One wave32 per edge; each lane handles 8 coalesced columns.
__global__ void edge_scatter_kernel(const float* __restrict__ x,
                                    const long long* __restrict__ ei,  // int64 [2,E]
                                    const float* __restrict__ w,
                                    float* __restrict__ agg,
                                    float* __restrict__ wsum) {
    int gid  = blockIdx.x * blockDim.x + threadIdx.x;
    int e    = gid >> 5;
    int lane = gid & 31;
    if (e >= EE) return;
    long long s = ei[e];           // src
    long long d = ei[EE + e];      // dst
    float we = w[e];
    const float* __restrict__ xr = x + (size_t)s * HH;
    float* __restrict__ ar = agg + (size_t)d * HH;
#pragma unroll
    for (int i = 0; i < 8; ++i) {
        int c = lane + 32 * i;     // lanes contiguous -> coalesced row reads
        atomicAdd(&ar[c], xr[c] * we);
    }
    if (lane == 0) atomicAdd(&wsum[d], we);
}

__device__ __forceinline__ float gelu_erf(float v) {
    return 0.5f * v * (1.0f + erff(v * 0.70710678118654752f));
}

// ---------------- Kernel 3: fused  (agg/wsum) -> GEMM1 -> GELU -> GEMM2
//                  -> +bias +residual -> LayerNorm -> out ----------------
// 512 threads = 16 waves. Wave w owns output columns [16w, 16w+16).
__global__ void __launch_bounds__(512)
fused_mlp_ln_kernel(const float* __restrict__ agg,
                    const float* __restrict__ wsum,
                    const float* __restrict__ x,
                    const float* __restrict__ W1, const float* __restrict__ b1,
                    const float* __restrict__ W2, const float* __restrict__ b2,
                    const float* __restrict__ gamma, const float* __restrict__ beta,
                    float* __restrict__ out) {
    __shared__ float aT[MT][LDP];   // agg tile, later reused as y tile
    __shared__ float hT[MT][LDP];   // hidden (GELU) tile

    const int tid     = threadIdx.x;
    const int lane    = tid & 31;
    const int waveId  = tid >> 5;
    const int rowBase = blockIdx.x * MT;

    // --- cooperative load of scaled agg tile (weighted mean) ---
    for (int idx = tid; idx < MT * HH; idx += 512) {
        int r = idx >> 8;          // /HH
        int c = idx & (HH - 1);
        float inv = 1.0f / fmaxf(wsum[rowBase + r], 1e-6f);
        aT[r][c] = agg[(size_t)(rowBase + r) * HH + c] * inv;
    }
    __syncthreads();

    // WMMA fragment indexing (fp32 16x16x4):
    //  A: lane -> row m = lane&15, K pair base kb = 2*(lane>>4)
    //  B (= W^T): same pattern on weight row n
    const int m  = lane & 15;
    const int kb = (lane >> 4) << 1;
    const int ng = waveId * 16 + m;          // this lane's B row / output col

    // ---------------- GEMM1: h = GELU(aggTile @ W1^T + b1) ----------------
    {
        const float* __restrict__ wrow = W1 + (size_t)ng * HH;
        v8f acc = {};
#pragma unroll 8
        for (int k = 0; k < HH; k += 4) {
            v2f a = *(const v2f*)&aT[m][k + kb];
            v2f b = *(const v2f*)&wrow[k + kb];
            acc = __builtin_amdgcn_wmma_f32_16x16x4_f32(
                false, a, false, b, (short)0, acc, false, false);
        }
        float bias = b1[ng];
#pragma unroll
        for (int r = 0; r < 8; ++r) {
            int mr = r + ((lane >> 4) << 3);   // C/D layout: VGPR r holds M=r (lanes 0-15), M=r+8 (16-31)
            hT[mr][ng] = gelu_erf(acc[r] + bias);
        }
    }
    __syncthreads();

    // ---------------- GEMM2: y = hTile @ W2^T + b2 + x  (into aT) ----------------
    {
        const float* __restrict__ wrow = W2 + (size_t)ng * HH;
        v8f acc = {};
#pragma unroll 8
        for (int k = 0; k < HH; k += 4) {
            v2f a = *(const v2f*)&hT[m][k + kb];
            v2f b = *(const v2f*)&wrow[k + kb];
            acc = __builtin_amdgcn_wmma_f32_16x16x4_f32(
                false, a, false, b, (short)0, acc, false, false);
        }
        float bias = b2[ng];
#pragma unroll
        for (int r = 0; r < 8; ++r) {
            int mr = r + ((lane >> 4) << 3);
            float resid = x[(size_t)(rowBase + mr) * HH + ng];
            aT[mr][ng] = acc[r] + bias + resid;
        }
    }
    __syncthreads();

    // ---------------- LayerNorm: wave w normalizes row w ----------------
    {
        const int row = waveId;                // 16 waves <-> 16 rows
        float sum = 0.0f, ssq = 0.0f;
#pragma unroll
        for (int j = 0; j < 8; ++j) {
            float v = aT[row][lane + 32 * j];
            sum += v;
            ssq += v * v;
        }
#pragma unroll
        for (int off = 16; off > 0; off >>= 1) {
            sum += __shfl_xor(sum, off, 32);
            ssq += __shfl_xor(ssq, off, 32);
        }
        float mu = sum * (1.0f / HH);
        float var = ssq * (1.0f / HH) - mu * mu;
        float rs = rsqrtf(var + 1e-5f);
        float* __restrict__ orow = out + (size_t)(rowBase + row) * HH;
#pragma unroll
        for (int j = 0; j < 8; ++j) {
            int c = lane + 32 * j;
            orow[c] = (aT[row][c] - mu) * rs * gamma[c] + beta[c];
        }
    }
}

extern "C" void kernel_launch(void* const* d_in, const int* in_sizes, int n_in,
                              void* d_out, int out_size, void* d_ws, size_t ws_size,
                              hipStream_t stream) {
    const float*     x     = (const float*)d_in[0];
    const long long* ei    = (const long long*)d_in[1];  // int64 per reference
    const float*     attr  = (const float*)d_in[2];
    const float*     W1    = (const float*)d_in[3];
    const float*     b1    = (const float*)d_in[4];
    const float*     W2    = (const float*)d_in[5];
    const float*     b2    = (const float*)d_in[6];
    const float*     gamma = (const float*)d_in[7];
    const float*     beta  = (const float*)d_in[8];
    float* out = (float*)d_out;

    // workspace layout: [agg N*H][wsum N][w E]
    float* agg  = (float*)d_ws;
    float* wsum = agg + (size_t)NN * HH;
    float* w    = wsum + NN;

    hipMemsetAsync(agg, 0, (size_t)(NN * HH + NN) * sizeof(float), stream);

    edge_weight_kernel<<<(EE + 255) / 256, 256, 0, stream>>>(attr, w);

    // one wave32 per edge
    long long scatter_threads = (long long)EE * 32;
    edge_scatter_kernel<<<(int)((scatter_threads + 255) / 256), 256, 0, stream>>>(
        x, ei, w, agg, wsum);

    fused_mlp_ln_kernel<<<NN / MT, 512, 0, stream>>>(
        agg, wsum, x, W1, b1, W2, b2, gamma, beta, out);
}